// DCFormer_66365834657868
// MI455X (gfx1250) — compile-verified
//
#include <hip/hip_runtime.h>
#include <math.h>

// ---- problem constants (B=1) ----
#define T_   2048
#define NH   16
#define DH   64
#define DD_  1024
#define CC   4
#define KK   128
#define QP   68      // padded LDS row stride (floats) for k/v rows of 64 -> bank-conflict-free b64 frag loads
#define LP   17      // padded head stride in logits tile [s][t][n]
#define PP_  18      // padded s stride in n-major P tile [n][t][s]
#define QN   (16*16*QP)   // 17408 floats per staged 16-head x 16-row panel
#define LN   (16*16*LP)   // 4352 floats logits tile
#define PN   (16*16*PP_)  // 4608 floats P tile

typedef float v2f __attribute__((ext_vector_type(2)));
typedef float v8f __attribute__((ext_vector_type(8)));

__device__ __forceinline__ v8f wmma_f32_16x16x4(v2f a, v2f b, v8f c) {
  // D = A(16x4 f32) * B(4x16 f32) + C(16x16 f32)
  return __builtin_amdgcn_wmma_f32_16x16x4_f32(false, a, false, b, (short)0, c, false, false);
}

// =====================================================================
// Kernel 1: per-t dynamic weights
// =====================================================================
__global__ __launch_bounds__(256) void weights_kernel(
    const float* __restrict__ x, const float* __restrict__ dw1,
    const float* __restrict__ qkw, const float* __restrict__ dd,
    float* __restrict__ w1n, float* __restrict__ w2o, float* __restrict__ ddo)
{
  extern __shared__ float sm[];
  float* xl = sm;                 // 16*1024
  float* hl = xl + 16*DD_;        // 16*512
  float* wl = hl + 16*512;        // 4096
  const int tid = threadIdx.x;
  const int tb  = blockIdx.x * 16;

  for (int i = tid; i < 16*DD_/4; i += 256)
    ((float4*)xl)[i] = ((const float4*)(x + (size_t)tb*DD_))[i];
  __syncthreads();

  // ddo = tanh(x @ dd)
  for (int o = tid; o < 16*64; o += 256) {
    int t = o >> 6, j = o & 63;
    float acc = 0.f;
    const float* xr = xl + t*DD_;
    for (int d = 0; d < DD_; ++d) acc += xr[d] * dd[d*64 + j];
    ddo[(size_t)(tb + t)*64 + j] = tanhf(acc);
  }

  // hidden = gelu(x @ dw1)
  {
    float acc0[16], acc1[16];
#pragma unroll
    for (int t = 0; t < 16; ++t) { acc0[t] = 0.f; acc1[t] = 0.f; }
    const int c0 = tid, c1 = tid + 256;
    for (int d = 0; d < DD_; ++d) {
      float w0 = dw1[(size_t)d*512 + c0];
      float w1 = dw1[(size_t)d*512 + c1];
#pragma unroll
      for (int t = 0; t < 16; ++t) {
        float xv = xl[t*DD_ + d];
        acc0[t] += xv * w0;
        acc1[t] += xv * w1;
      }
    }
#pragma unroll
    for (int t = 0; t < 16; ++t) {
      float a = acc0[t];
      hl[t*512 + c0] = 0.5f*a*(1.f + erff(a*0.70710678118f));
      a = acc1[t];
      hl[t*512 + c1] = 0.5f*a*(1.f + erff(a*0.70710678118f));
    }
  }
  __syncthreads();

  // w[t][c][i][m] = sum_k hidden[t][c*128+k] * qkw[c][k][i][m]
  for (int o = tid; o < 4096; o += 256) {
    int t = o >> 8, r = o & 255, c = r >> 6, im = r & 63;
    float acc = 0.f;
    const float* hr = hl + t*512 + c*KK;
    const float* qw = qkw + (size_t)c*KK*64 + im;
    for (int kk = 0; kk < KK; ++kk) acc += hr[kk] * qw[(size_t)kk*64];
    wl[o] = acc;
  }
  __syncthreads();

  // RMS-norm w1 (i=0,1) over m; copy w2 (i=2,3)
  if (tid < 128) {
    int t = tid >> 3, r = tid & 7, c = r >> 1, i = r & 1;
    const float* g1 = wl + ((t*4 + c)*4 + i)*16;
    const float* g2 = wl + ((t*4 + c)*4 + i + 2)*16;
    float ms = 0.f;
#pragma unroll
    for (int m = 0; m < 16; ++m) ms += g1[m]*g1[m];
    float sc = rsqrtf(ms * (1.f/16.f) + 1e-6f);
    float* o1 = w1n + (size_t)(((tb + t)*4 + c)*2 + i)*16;
    float* o2 = w2o + (size_t)(((tb + t)*4 + c)*2 + i)*16;
#pragma unroll
    for (int m = 0; m < 16; ++m) { o1[m] = g1[m]*sc; o2[m] = g2[m]; }
  }
}

// =====================================================================
// Shared helpers for attention passes
// =====================================================================

__device__ __forceinline__ void load_rows(const float* __restrict__ g, int base,
                                          float* __restrict__ l, int tid)
{
  for (int i = tid; i < 4096; i += 256) {
    int n   = i >> 8;
    int row = (i >> 4) & 15;
    int d4  = (i & 15) << 2;
    float4 v = *(const float4*)(g + ((size_t)(n*T_ + base + row))*64 + d4);
    float* dst = l + (n*16 + row)*QP + d4;
    dst[0]=v.x; dst[1]=v.y; dst[2]=v.z; dst[3]=v.w;
  }
}

// [0]=pre_w1 [512]=pre_w2 [1024]=post_w1 [1536]=post_w2 [2048]=pre_dd [2304]=post_dd
__device__ __forceinline__ void load_side_weights(
    const float* __restrict__ w1n, const float* __restrict__ w2o,
    const float* __restrict__ ddo, int base, int cPre, int cPost,
    int ddPre, int ddPost, float* __restrict__ dst, int tid)
{
  for (int i = tid; i < 512; i += 256) {
    int t = i >> 5, im = i & 31;
    dst[i]        = w1n[(size_t)((base + t)*4 + cPre )*32 + im];
    dst[512 + i]  = w2o[(size_t)((base + t)*4 + cPre )*32 + im];
    dst[1024 + i] = w1n[(size_t)((base + t)*4 + cPost)*32 + im];
    dst[1536 + i] = w2o[(size_t)((base + t)*4 + cPost)*32 + im];
  }
  {
    int t = tid >> 4, m = tid & 15;
    dst[2048 + tid] = ddo[(size_t)(base + t)*64 + ddPre  + m];
    dst[2304 + tid] = ddo[(size_t)(base + t)*64 + ddPost + m];
  }
}

// Load loop-invariant A fragments (q) for this wave's 2 heads, direct from global.
__device__ __forceinline__ void load_q_frags(const float* __restrict__ q, int tb,
                                             int wave, int lane, v2f (&qa)[2][16])
{
  const int m   = lane & 15;
  const int kb2 = (lane >> 4) << 1;
#pragma unroll
  for (int j = 0; j < 2; ++j) {
    const float* qrow = q + ((size_t)((wave*2 + j)*T_ + tb + m))*64 + kb2;
#pragma unroll
    for (int kk = 0; kk < 16; ++kk)
      qa[j][kk] = *(const v2f*)(qrow + kk*4);
  }
}

// logits tile: wave owns heads {2w, 2w+1}; C(16t x 16s) = q(16x64) @ k^T(64x16)
__device__ __forceinline__ void logits_tiles(const v2f (&qa)[2][16],
                                             const float* __restrict__ kl,
                                             float* __restrict__ Ll,
                                             int wave, int lane)
{
  const int m   = lane & 15;          // B: s col (and A t row baked into qa)
  const int kb2 = (lane >> 4) << 1;
#pragma unroll
  for (int j = 0; j < 2; ++j) {
    const int head = wave*2 + j;
    v8f c = {};
    const float* krow = kl + (head*16 + m)*QP + kb2;
#pragma unroll
    for (int kk = 0; kk < 16; ++kk) {
      v2f b = *(const v2f*)(krow + kk*4);
      c = wmma_f32_16x16x4(qa[j][kk], b, c);
    }
    const int sl = lane & 15;
    const int hb = (lane >> 4) << 3;
#pragma unroll
    for (int r = 0; r < 8; ++r)
      Ll[(sl*16 + r + hb)*LP + head] = c[r];   // [s][t][n]
  }
}

// qh/kh per (t,s): thread = s*16 + t
__device__ __forceinline__ void qkh_stage(const float* __restrict__ Ll,
                                          float* __restrict__ qkh,
                                          const float* __restrict__ tw1,
                                          const float* __restrict__ sw1, int tid)
{
  const int s = tid >> 4, t = tid & 15;
  const float* Lp = Ll + (s*16 + t)*LP;
  const float* tq = tw1 + t*32;
  const float* sk = sw1 + s*32;
  float q0=0.f,q1=0.f,k0=0.f,k1=0.f;
#pragma unroll
  for (int m = 0; m < 16; ++m) {
    float v = Lp[m];
    q0 += v * tq[m];  q1 += v * tq[16+m];
    k0 += v * sk[m];  k1 += v * sk[16+m];
  }
  float* o = qkh + tid*4;
  o[0]=q0; o[1]=q1; o[2]=k0; o[3]=k1;
}

__device__ __forceinline__ float combine_one(float v, const float* __restrict__ qk4,
    float qw2a, float qw2b, float kw2a, float kw2b, float qddv, float kddv)
{
  return v*(1.f + qddv + kddv) + qk4[0]*qw2a + qk4[1]*qw2b + qk4[2]*kw2a + qk4[3]*kw2b;
}

// =====================================================================
// Pass 2: online softmax stats of cross_pre(q@k^T) per (head,t) row
// LDS ~112KB -> 2 workgroups / WGP
// =====================================================================
__global__ __launch_bounds__(256) void pass2_kernel(
    const float* __restrict__ q, const float* __restrict__ k,
    const float* __restrict__ w1n, const float* __restrict__ w2o,
    const float* __restrict__ ddo,
    float* __restrict__ rowmax, float* __restrict__ rowsum)
{
  extern __shared__ float sm[];
  float* kl  = sm;
  float* Ll  = kl + QN;
  float* qkh = Ll + LN;
  float* tw  = qkh + 1024;
  float* sw  = tw + 2560;

  const int tid  = threadIdx.x;
  const int wave = tid >> 5, lane = tid & 31;
  const int tb   = blockIdx.x * 16;
  const int n = tid & 15, tloc = tid >> 4;

  v2f qa[2][16];
  load_q_frags(q, tb, wave, lane, qa);
  load_side_weights(w1n, w2o, ddo, tb, 0, 2, 0, 32, tw, tid);   // t-side: pre_q, post_q

  float mrun = -3.0e38f, srun = 0.f;

  for (int sc = 0; sc < T_/16; ++sc) {
    __syncthreads();
    load_rows(k, sc*16, kl, tid);
    load_side_weights(w1n, w2o, ddo, sc*16, 1, 3, 16, 48, sw, tid);  // s-side: pre_k, post_k
    __syncthreads();
    logits_tiles(qa, kl, Ll, wave, lane);
    __syncthreads();
    qkh_stage(Ll, qkh, tw, sw, tid);
    __syncthreads();

    const float tq2a = tw[512 + tloc*32 + n];
    const float tq2b = tw[512 + tloc*32 + 16 + n];
    const float tdd  = tw[2048 + tloc*16 + n];
    for (int s2 = 0; s2 < 16; ++s2) {
      float v = Ll[(s2*16 + tloc)*LP + n];
      float L = combine_one(v, qkh + (s2*16 + tloc)*4, tq2a, tq2b,
                            sw[512 + s2*32 + n], sw[512 + s2*32 + 16 + n],
                            tdd, sw[2048 + s2*16 + n]);
      float nm = fmaxf(mrun, L);
      srun = srun * expf(mrun - nm) + expf(L - nm);
      mrun = nm;
    }
  }
  rowmax[n*T_ + tb + tloc] = mrun;
  rowsum[n*T_ + tb + tloc] = srun;
}

// =====================================================================
// Pass 3: recompute, normalize, cross_post, O += P@V
// =====================================================================
__global__ __launch_bounds__(256) void pass3_kernel(
    const float* __restrict__ q, const float* __restrict__ k,
    const float* __restrict__ v,
    const float* __restrict__ w1n, const float* __restrict__ w2o,
    const float* __restrict__ ddo,
    const float* __restrict__ rowmax, const float* __restrict__ rowsum,
    float* __restrict__ out)
{
  extern __shared__ float sm[];
  float* kl  = sm;
  float* vl  = kl + QN;
  float* Ll  = vl + QN;
  float* Pl  = Ll + LN;     // [n][t][PP_] n-major post-projected probs
  float* qkh = Pl + PN;
  float* tw  = qkh + 1024;
  float* sw  = tw + 2560;

  const int tid  = threadIdx.x;
  const int wave = tid >> 5, lane = tid & 31;
  const int tb   = blockIdx.x * 16;
  const int n = tid & 15, tloc = tid >> 4;

  v2f qa[2][16];
  load_q_frags(q, tb, wave, lane, qa);
  load_side_weights(w1n, w2o, ddo, tb, 0, 2, 0, 32, tw, tid);

  const float mrow = rowmax[n*T_ + tb + tloc];
  const float inv  = 1.f / rowsum[n*T_ + tb + tloc];

  v8f zero = {};
  v8f acc[2][4];
#pragma unroll
  for (int j = 0; j < 2; ++j)
#pragma unroll
    for (int dt = 0; dt < 4; ++dt) acc[j][dt] = zero;

  for (int sc = 0; sc < T_/16; ++sc) {
    __syncthreads();
    load_rows(k, sc*16, kl, tid);
    load_rows(v, sc*16, vl, tid);
    load_side_weights(w1n, w2o, ddo, sc*16, 1, 3, 16, 48, sw, tid);
    __syncthreads();
    logits_tiles(qa, kl, Ll, wave, lane);
    __syncthreads();
    qkh_stage(Ll, qkh, tw, sw, tid);                 // pre weights
    __syncthreads();
    {  // cross_pre + softmax normalize, in place in Ll
      const float tq2a = tw[512 + tloc*32 + n];
      const float tq2b = tw[512 + tloc*32 + 16 + n];
      const float tdd  = tw[2048 + tloc*16 + n];
      for (int s2 = 0; s2 < 16; ++s2) {
        float val = Ll[(s2*16 + tloc)*LP + n];
        float L = combine_one(val, qkh + (s2*16 + tloc)*4, tq2a, tq2b,
                              sw[512 + s2*32 + n], sw[512 + s2*32 + 16 + n],
                              tdd, sw[2048 + s2*16 + n]);
        Ll[(s2*16 + tloc)*LP + n] = expf(L - mrow) * inv;
      }
    }
    __syncthreads();
    qkh_stage(Ll, qkh, tw + 1024, sw + 1024, tid);   // post weights
    __syncthreads();
    {  // cross_post -> n-major P tile
      const float tq2a = tw[1536 + tloc*32 + n];
      const float tq2b = tw[1536 + tloc*32 + 16 + n];
      const float tdd  = tw[2304 + tloc*16 + n];
      float* prow = Pl + (n*16 + tloc)*PP_;
      for (int s2 = 0; s2 < 16; ++s2) {
        float p = Ll[(s2*16 + tloc)*LP + n];
        prow[s2] = combine_one(p, qkh + (s2*16 + tloc)*4, tq2a, tq2b,
                               sw[1536 + s2*32 + n], sw[1536 + s2*32 + 16 + n],
                               tdd, sw[2304 + s2*16 + n]);
      }
    }
    __syncthreads();
    {  // O(16t x 64d) += P(16t x 16s) @ V(16s x 64d), per owned head
      const int ml  = lane & 15;
      const int kb2 = (lane >> 4) << 1;
#pragma unroll
      for (int j = 0; j < 2; ++j) {
        const int head = wave*2 + j;
        const float* prow = Pl + (head*16 + ml)*PP_ + kb2;
#pragma unroll
        for (int dt = 0; dt < 4; ++dt) {
          v8f c = acc[j][dt];
#pragma unroll
          for (int kk = 0; kk < 4; ++kk) {
            const int s0 = kk*4 + kb2;
            v2f a = *(const v2f*)(prow + kk*4);
            v2f b;
            b.x = vl[(head*16 + s0)*QP + dt*16 + ml];
            b.y = vl[(head*16 + s0 + 1)*QP + dt*16 + ml];
            c = wmma_f32_16x16x4(a, b, c);
          }
          acc[j][dt] = c;
        }
      }
    }
  }

  // write O: element (t = r + 8*(lane>>4), d = dt*16 + lane&15)
  const int ml = lane & 15, hb = (lane >> 4) << 3;
#pragma unroll
  for (int j = 0; j < 2; ++j) {
    const int head = wave*2 + j;
#pragma unroll
    for (int dt = 0; dt < 4; ++dt)
#pragma unroll
      for (int r = 0; r < 8; ++r)
        out[((size_t)(head*T_) + tb + r + hb)*64 + dt*16 + ml] = acc[j][dt][r];
  }
}

// =====================================================================
extern "C" void kernel_launch(void* const* d_in, const int* in_sizes, int n_in,
                              void* d_out, int out_size, void* d_ws, size_t ws_size,
                              hipStream_t stream) {
  const float* q   = (const float*)d_in[0];
  const float* k   = (const float*)d_in[1];
  const float* v   = (const float*)d_in[2];
  const float* x   = (const float*)d_in[3];
  const float* dw1 = (const float*)d_in[4];
  const float* qkw = (const float*)d_in[5];
  const float* dd  = (const float*)d_in[6];
  float* out = (float*)d_out;

  // workspace layout (floats)
  float* ws     = (float*)d_ws;
  float* w1n    = ws;                       // [T][4][2][16]  = 262144
  float* w2o    = w1n + 262144;             // [T][4][2][16]  = 262144
  float* ddo    = w2o + 262144;             // [T][64]        = 131072
  float* rowmax = ddo + 131072;             // [N][T]         = 32768
  float* rowsum = rowmax + 32768;           // [N][T]         = 32768

  const int lds_w = (16*DD_ + 16*512 + 4096) * 4;                   // 114688
  const int lds_2 = (QN + LN + 1024 + 2560 + 2560) * 4;             // 111616
  const int lds_3 = (2*QN + LN + PN + 1024 + 2560 + 2560) * 4;      // 199680

  (void)hipFuncSetAttribute((const void*)weights_kernel,
      hipFuncAttributeMaxDynamicSharedMemorySize, lds_w);
  (void)hipFuncSetAttribute((const void*)pass2_kernel,
      hipFuncAttributeMaxDynamicSharedMemorySize, lds_2);
  (void)hipFuncSetAttribute((const void*)pass3_kernel,
      hipFuncAttributeMaxDynamicSharedMemorySize, lds_3);

  weights_kernel<<<T_/16, 256, lds_w, stream>>>(x, dw1, qkw, dd, w1n, w2o, ddo);
  pass2_kernel<<<T_/16, 256, lds_2, stream>>>(q, k, w1n, w2o, ddo, rowmax, rowsum);
  pass3_kernel<<<T_/16, 256, lds_3, stream>>>(q, k, v, w1n, w2o, ddo, rowmax, rowsum, out);
}